// PureTriadicBrain_45311904972977
// MI455X (gfx1250) — compile-verified
//
#include <hip/hip_runtime.h>
#include <math.h>
#include <stdint.h>

// ---------------- problem constants ----------------
#define B_   4
#define N_   256
#define D_   128
#define BN_  1024            // B*N
#define NN_  65536           // N*N
#define DD_  16384           // D*D

// model hyper-params
#define EMA_SPEED   0.05f
#define LR_         0.006f
#define MOMENTUM_   0.4f
#define SPEED_      0.5f
#define WD_         0.01f
#define NOISE_SC_   0.8f
#define C_MOMENT_   0.99f
#define BOUND_      1.8477590650225735f   // sqrt(2+sqrt(2))
#define UNTANH_     0.6f
// W_SCALE * DYN_SCALE = 0.27 / (BOUND*|1-UNTANH| + 1e-6)
#define WSCALE_DYN_ (0.27f / (BOUND_ * 0.4f + 1e-6f))
#define INV_SQRT_D_ 0.08838834764831845f  // 1/sqrt(128)
#define UB_         (1.0f / 65537.0f)     // 1/(N*N+1)

typedef float v2f __attribute__((ext_vector_type(2)));
typedef float v8f __attribute__((ext_vector_type(8)));

#if defined(__AMDGCN__) && __has_builtin(__builtin_amdgcn_wmma_f32_16x16x4_f32)
#define HAVE_WMMA_F32 1
#else
#define HAVE_WMMA_F32 0
#endif

// ---------------- small helpers ----------------
__device__ __forceinline__ float blockSum(float v, float* red) {
  for (int o = 16; o > 0; o >>= 1) v += __shfl_down(v, o);
  int lane = threadIdx.x & 31, w = threadIdx.x >> 5, nw = blockDim.x >> 5;
  if (lane == 0) red[w] = v;
  __syncthreads();
  float s = 0.f;
  for (int k = 0; k < nw; ++k) s += red[k];
  __syncthreads();
  return s;
}

__device__ __forceinline__ float blockMax(float v, float* red) {
  for (int o = 16; o > 0; o >>= 1) v = fmaxf(v, __shfl_down(v, o));
  int lane = threadIdx.x & 31, w = threadIdx.x >> 5, nw = blockDim.x >> 5;
  if (lane == 0) red[w] = v;
  __syncthreads();
  float s = -3.0e38f;
  for (int k = 0; k < nw; ++k) s = fmaxf(s, red[k]);
  __syncthreads();
  return s;
}

__device__ __forceinline__ uint32_t hash_u32(uint32_t x) {
  x ^= x >> 16; x *= 0x7feb352dU;
  x ^= x >> 15; x *= 0x846ca68bU;
  x ^= x >> 16; return x;
}

__device__ __forceinline__ float gauss_rand(uint32_t gsel, uint32_t idx) {
  uint32_t h1 = hash_u32(idx * 3u + gsel + 0x68bc21ebU);
  uint32_t h2 = hash_u32(h1 ^ (0x02e5be93U + gsel * 0x9e3779b9U));
  float u1 = ((float)h1 + 0.5f) * 2.3283064365386963e-10f;
  float u2 = ((float)h2 + 0.5f) * 2.3283064365386963e-10f;
  return sqrtf(-2.f * __logf(u1)) * __cosf(6.2831853071795864f * u2);
}

// ---------------- K0: zero small ws region ----------------
__global__ void k_zero(float* p, int n) {
  int i = blockIdx.x * blockDim.x + threadIdx.x;
  if (i < n) p[i] = 0.f;
}

// ---------------- K1: batched GEMV + buntanh ----------------
// blockIdx.x = bn, blockIdx.y = which (0:w1->pred, 1:w2->keys, 2:w3->queries)
__global__ void k_matvec(const float* __restrict__ state,
                         const float* __restrict__ w1,
                         const float* __restrict__ w2,
                         const float* __restrict__ w3,
                         float* __restrict__ pred_out,
                         float* __restrict__ kbuf,
                         float* __restrict__ qbuf) {
  int bn = blockIdx.x;
  int which = blockIdx.y;
  const float* w = (which == 0) ? w1 : ((which == 1) ? w2 : w3);
  float* dst = (which == 0) ? pred_out : ((which == 1) ? kbuf : qbuf);
  __shared__ float s[D_];
  int t = threadIdx.x;  // 0..127
  s[t] = state[bn * D_ + t];
  __syncthreads();
  const float* wm = w + (size_t)bn * DD_;
  float acc = 0.f;
#pragma unroll 8
  for (int j = 0; j < D_; ++j) acc += s[j] * wm[j * D_ + t];  // coalesced across t
  // buntanh(x) = ((1-u)*tanh(x*B) + u*x*B)/B
  float x = acc * BOUND_;
  float r = ((1.f - UNTANH_) * tanhf(x) + UNTANH_ * x) * (1.0f / BOUND_);
  dst[bn * D_ + t] = r;
}

// ---------------- K2a: raw_A = Q*K^T / sqrt(D), per-batch sumsq ----------------
// 1 wave = one 16x16 output tile.  grid = 128 blocks x 256 threads (8 waves)
__global__ void k_qk_gemm(const float* __restrict__ qbuf,
                          const float* __restrict__ kbuf,
                          float* __restrict__ raw,
                          float* __restrict__ sumsq) {
  int tid = threadIdx.x;
  int lane = tid & 31, wid = tid >> 5;
  int tile = blockIdx.x * 8 + wid;       // 0..1023
  int b = tile >> 8;                     // 256 tiles per batch
  int t = tile & 255;
  int tm = t >> 4, tn = t & 15;
  const float* Qt = qbuf + (size_t)b * N_ * D_ + tm * 16 * D_;
  const float* Kt = kbuf + (size_t)b * N_ * D_ + tn * 16 * D_;
  int row = lane & 15;   // A: M row / B: N col
  int hi  = lane >> 4;   // K-pair select
  float cvals[8];
#if HAVE_WMMA_F32
  v8f c;
#pragma unroll
  for (int z = 0; z < 8; ++z) c[z] = 0.f;
  for (int kk = 0; kk < D_; kk += 4) {
    v2f a, bb;
    a.x  = Qt[row * D_ + kk + 2 * hi];
    a.y  = Qt[row * D_ + kk + 1 + 2 * hi];
    bb.x = Kt[row * D_ + kk + 2 * hi];
    bb.y = Kt[row * D_ + kk + 1 + 2 * hi];
    c = __builtin_amdgcn_wmma_f32_16x16x4_f32(false, a, false, bb,
                                              (short)0, c, false, false);
  }
#pragma unroll
  for (int v = 0; v < 8; ++v) cvals[v] = c[v];
#else
#pragma unroll
  for (int v = 0; v < 8; ++v) {
    int ml = v + 8 * hi;
    float acc = 0.f;
    for (int k = 0; k < D_; ++k) acc += Qt[ml * D_ + k] * Kt[row * D_ + k];
    cvals[v] = acc;
  }
#endif
  float* outp = raw + (size_t)b * NN_;
  float local = 0.f;
#pragma unroll
  for (int v = 0; v < 8; ++v) {
    int m = tm * 16 + v + 8 * hi;
    int n = tn * 16 + row;
    float val = cvals[v] * INV_SQRT_D_;
    outp[m * N_ + n] = val;
    local += val * val;
  }
  for (int o = 16; o > 0; o >>= 1) local += __shfl_down(local, o);
  if (lane == 0) atomicAdd(&sumsq[b], local);
}

// ---------------- K2b: rms_norm over (N,N) + EMA -> a_ema_new ----------------
__global__ void k_norm_ema(const float* __restrict__ raw,
                           const float* __restrict__ sumsq,
                           const float* __restrict__ aema_in,
                           const int* __restrict__ stepc,
                           float* __restrict__ aema_out) {
  int idx = blockIdx.x * blockDim.x + threadIdx.x;
  if (idx >= B_ * NN_) return;
  int b = idx >> 16;
  float mean = sumsq[b] * (1.f / (float)NN_);
  float val = raw[idx] * rsqrtf(mean + 1e-8f);
  int sc = stepc[0] + 1;
  float out = (sc == 1) ? val : (aema_in[idx] * C_MOMENT_ + val * (1.f - C_MOMENT_));
  aema_out[idx] = out;
}

// ---------------- K3: double Sinkhorn + sparsify + row-zero, fully in LDS ----
#define SKP 257  // stride-pad: bank(r*257+j)%64 = (r+j)%64 -> conflict-free both ways
__global__ void k_sinkhorn(const float* __restrict__ aema, float* __restrict__ Aout) {
  extern __shared__ float sm[];  // 256*257 floats
  int b = blockIdx.x;
  int t = threadIdx.x;  // 256 threads
  const float* src = aema + (size_t)b * NN_;
  for (int r = 0; r < N_; ++r) sm[r * SKP + t] = src[r * N_ + t];
  __syncthreads();
  for (int phase = 0; phase < 2; ++phase) {
    for (int it = 0; it < 10; ++it) {
      // rows (axis=-1): thread t owns row t
      float m = -3.0e38f;
      for (int j = 0; j < N_; ++j) m = fmaxf(m, sm[t * SKP + j]);
      float s = 0.f;
      for (int j = 0; j < N_; ++j) s += __expf(sm[t * SKP + j] - m);
      float lse = m + __logf(s);
      for (int j = 0; j < N_; ++j) sm[t * SKP + j] -= lse;
      __syncthreads();
      // cols (axis=-2): thread t owns column t
      m = -3.0e38f;
      for (int i = 0; i < N_; ++i) m = fmaxf(m, sm[i * SKP + t]);
      s = 0.f;
      for (int i = 0; i < N_; ++i) s += __expf(sm[i * SKP + t] - m);
      lse = m + __logf(s);
      for (int i = 0; i < N_; ++i) sm[i * SKP + t] -= lse;
      __syncthreads();
    }
    for (int r = 0; r < N_; ++r) sm[r * SKP + t] = __expf(sm[r * SKP + t]);
    __syncthreads();
    if (phase == 0) {  // sparsify: probs become next logits
      for (int r = 0; r < N_; ++r) {
        float v = sm[r * SKP + t];
        sm[r * SKP + t] = (v > UB_) ? v : 0.f;
      }
      __syncthreads();
    }
  }
  float* dst = Aout + (size_t)b * NN_;
  for (int r = 0; r < N_; ++r) dst[r * N_ + t] = (r < 2) ? 0.f : sm[r * SKP + t];
}

// ---------------- K4: target = A * output (+ env-row substitution) ----------
// 512 tiles / 8 waves = 64 blocks x 256 threads
__global__ void k_target(const float* __restrict__ Amat,
                         const float* __restrict__ outp,
                         const float* __restrict__ eye,
                         const float* __restrict__ stomach,
                         float* __restrict__ nstate) {
  int tid = threadIdx.x;
  int lane = tid & 31, wid = tid >> 5;
  int tile = blockIdx.x * 8 + wid;   // 0..511
  int b = tile >> 7;                 // 128 tiles per batch (16 m-tiles x 8 d-tiles)
  int t = tile & 127;
  int tm = t >> 3, tn = t & 7;
  const float* At = Amat + (size_t)b * NN_ + tm * 16 * N_;
  const float* Ot = outp + (size_t)b * N_ * D_;
  int row = lane & 15;
  int hi  = lane >> 4;
  int dcol = tn * 16 + row;
  float cvals[8];
#if HAVE_WMMA_F32
  v8f c;
#pragma unroll
  for (int z = 0; z < 8; ++z) c[z] = 0.f;
  for (int kk = 0; kk < N_; kk += 4) {
    v2f a, bb;
    a.x  = At[row * N_ + kk + 2 * hi];
    a.y  = At[row * N_ + kk + 1 + 2 * hi];
    bb.x = Ot[(kk + 2 * hi) * D_ + dcol];
    bb.y = Ot[(kk + 1 + 2 * hi) * D_ + dcol];
    c = __builtin_amdgcn_wmma_f32_16x16x4_f32(false, a, false, bb,
                                              (short)0, c, false, false);
  }
#pragma unroll
  for (int v = 0; v < 8; ++v) cvals[v] = c[v];
#else
#pragma unroll
  for (int v = 0; v < 8; ++v) {
    int ml = v + 8 * hi;
    float acc = 0.f;
    for (int k = 0; k < N_; ++k) acc += At[ml * N_ + k] * Ot[k * D_ + dcol];
    cvals[v] = acc;
  }
#endif
  float* dst = nstate + (size_t)b * N_ * D_;
#pragma unroll
  for (int v = 0; v < 8; ++v) {
    int m = tm * 16 + v + 8 * hi;
    float val = cvals[v];
    if (m == 0) val = eye[b * D_ + dcol];
    else if (m == 1) val = stomach[b * D_ + dcol];
    dst[m * D_ + dcol] = val;
  }
}

// ---------------- K5: errors, softmax baselines, rank-1 gradient factors -----
// grad[i][j] = p_i * ns_j  (analytic collapse of rms_norm(outer(err,ns)))
__global__ void k_grad(const float* __restrict__ state,
                       const float* __restrict__ pred,
                       const float* __restrict__ targ,
                       const float* __restrict__ qbuf,
                       const float* __restrict__ kbuf,
                       const float* __restrict__ eb1,
                       const float* __restrict__ eb2,
                       const float* __restrict__ eb3,
                       float* __restrict__ eb1o,
                       float* __restrict__ eb2o,
                       float* __restrict__ eb3o,
                       float* __restrict__ pbuf,
                       float* __restrict__ nsbuf) {
  __shared__ float red[8];
  int bn = blockIdx.x;
  int t = threadIdx.x;  // 128
  int idx = bn * D_ + t;
  float pr = pred[idx], tg = targ[idx];
  float qv = qbuf[idx], kv = kbuf[idx], st = state[idx];
  float e1b = pr - tg;
  float e2 = kv - e1b;
  float e3 = qv - e2;
  float e1 = e1b - e3;
  float errs[3] = {e1, e2, e3};
  const float* ebis[3] = {eb1, eb2, eb3};
  float* ebos[3] = {eb1o, eb2o, eb3o};
#pragma unroll
  for (int g = 0; g < 3; ++g) {
    float err = errs[g];
    float m = blockMax(err, red);
    float ex = __expf(err - m);
    float s = blockSum(ex, red);
    float ec = ex / s;
    float ebn = (1.f - EMA_SPEED) * ebis[g][idx] + EMA_SPEED * ec;
    ebos[g][idx] = ebn;
    float adv = ec - ebn;
    float plast = 1.f + adv * rsqrtf(adv * adv + 1e-8f);
    float nz = gauss_rand((uint32_t)g, (uint32_t)idx);
    float ns = st + NOISE_SC_ * nz * err;
    float msq = blockSum(ns * ns, red) * (1.f / (float)D_);
    float p = plast * (-err) * rsqrtf(err * err * msq + 1e-8f);
    pbuf[g * (BN_ * D_) + idx] = p;
    nsbuf[g * (BN_ * D_) + idx] = ns;
  }
}

// ---------------- K6: piece_step, single pass, w_new in registers -----------
// grid = 1024 blocks x 256 threads; thread t owns row i=t>>1, cols (t&1)*64..+64
__global__ void k_piece(const float* __restrict__ w,
                        const float* __restrict__ ge,
                        const float* __restrict__ p,
                        const float* __restrict__ ns,
                        const int* __restrict__ stepc,
                        float* __restrict__ wout,
                        float* __restrict__ geout) {
  __shared__ float nskj[D_];
  __shared__ float kernj[D_];
  __shared__ float red[8];
  __shared__ float bcast;
  int bn = blockIdx.x;
  int t = threadIdx.x;  // 256
  int scv = stepc[0] + 1;
  if (t < D_) {
    float center = fmodf((float)scv * SPEED_, (float)D_);
    float d = fabsf((float)t - center);
    d = fminf(d, (float)D_ - d);
    float z = d * (1.0f / 12.8f);  // sigma = 0.1*D
    float kv = __expf(-0.5f * z * z);
    kernj[t] = kv;
    nskj[t] = ns[bn * D_ + t] * kv;
  }
  __syncthreads();
  int i = t >> 1;
  int jb = (t & 1) * 64;
  float pi = p[bn * D_ + i];
  size_t base = (size_t)bn * DD_ + (size_t)t * 64;
  const float4* w4 = (const float4*)(w + base);
  const float4* g4 = (const float4*)(ge + base);
  float4* go4 = (float4*)(geout + base);
  float wn[64];
  float acc = 0.f;
#pragma unroll
  for (int u = 0; u < 16; ++u) {
    float4 wv = w4[u];
    float4 gv = g4[u];
#pragma unroll
    for (int c2 = 0; c2 < 4; ++c2) {
      int j = jb + u * 4 + c2;
      float wc = (&wv.x)[c2];
      float gc = (&gv.x)[c2];
      // g = (p_i*ns_j - WD*w) * kern_j
      float gval = pi * nskj[j] - WD_ * wc * kernj[j];
      float gen = MOMENTUM_ * gc + (1.f - MOMENTUM_) * gval;
      (&gv.x)[c2] = gen;
      float wnv = wc + LR_ * gen;
      wn[u * 4 + c2] = wnv;
      acc += wnv * wnv;
    }
    go4[u] = gv;  // g_ema_new (pre-normalization, per reference)
  }
  // block reduce sum of squares over all 16384 elements
  for (int o = 16; o > 0; o >>= 1) acc += __shfl_down(acc, o);
  int lane = t & 31, wid = t >> 5;
  if (lane == 0) red[wid] = acc;
  __syncthreads();
  if (t == 0) {
    float s = 0.f;
    for (int k = 0; k < 8; ++k) s += red[k];
    bcast = WSCALE_DYN_ * rsqrtf(s * (1.f / (float)DD_) + 1e-8f);
  }
  __syncthreads();
  float scale = bcast;
  float4* wo4 = (float4*)(wout + base);
#pragma unroll
  for (int u = 0; u < 16; ++u) {
    float4 o;
    o.x = wn[u * 4 + 0] * scale;
    o.y = wn[u * 4 + 1] * scale;
    o.z = wn[u * 4 + 2] * scale;
    o.w = wn[u * 4 + 3] * scale;
    wo4[u] = o;
  }
}

// ---------------- launcher ----------------
extern "C" void kernel_launch(void* const* d_in, const int* in_sizes, int n_in,
                              void* d_out, int out_size, void* d_ws, size_t ws_size,
                              hipStream_t stream) {
  (void)in_sizes; (void)n_in; (void)out_size; (void)ws_size;
  const float* state   = (const float*)d_in[0];
  const float* output  = (const float*)d_in[1];
  const float* eye     = (const float*)d_in[2];
  const float* stomach = (const float*)d_in[3];
  const float* w1 = (const float*)d_in[4];
  const float* w2 = (const float*)d_in[5];
  const float* w3 = (const float*)d_in[6];
  const float* g1 = (const float*)d_in[7];
  const float* g2 = (const float*)d_in[8];
  const float* g3 = (const float*)d_in[9];
  const float* eb1 = (const float*)d_in[10];
  const float* eb2 = (const float*)d_in[11];
  const float* eb3 = (const float*)d_in[12];
  const float* aema = (const float*)d_in[13];
  const int* stepc = (const int*)d_in[14];

  float* out = (float*)d_out;
  // output tuple offsets (floats)
  float* o_state = out + 0;                          // (B,N,D)
  float* o_out   = out + 131072;                     // (B,N,D)
  float* o_A     = out + 262144;                     // (B,N,N)
  float* o_w1    = out + 524288;                     // (B,N,D,D)
  float* o_w2    = out + 524288 + 1 * 16777216;
  float* o_w3    = out + 524288 + 2 * 16777216;
  float* o_g1    = out + 524288 + 3 * 16777216;
  float* o_g2    = out + 524288 + 4 * 16777216;
  float* o_g3    = out + 524288 + 5 * 16777216;
  float* o_eb1   = out + 524288 + 6 * 16777216;      // 101187584
  float* o_eb2   = o_eb1 + 131072;
  float* o_eb3   = o_eb2 + 131072;
  float* o_aema  = o_eb3 + 131072;                   // (B,N,N)

  // workspace layout (floats): q, k, rawA, sumsq(pad 64), p[3], ns[3] ~= 5.3 MB
  float* ws = (float*)d_ws;
  float* ws_q   = ws;
  float* ws_k   = ws + 131072;
  float* ws_raw = ws + 262144;
  float* ws_ss  = ws + 524288;
  float* ws_p   = ws + 524352;
  float* ws_ns  = ws + 917568;

  k_zero<<<1, 64, 0, stream>>>(ws_ss, 64);

  dim3 gmv(BN_, 3);
  k_matvec<<<gmv, 128, 0, stream>>>(state, w1, w2, w3, o_out, ws_k, ws_q);

  k_qk_gemm<<<128, 256, 0, stream>>>(ws_q, ws_k, ws_raw, ws_ss);
  k_norm_ema<<<1024, 256, 0, stream>>>(ws_raw, ws_ss, aema, stepc, o_aema);

  k_sinkhorn<<<4, 256, N_ * SKP * sizeof(float), stream>>>(o_aema, o_A);

  k_target<<<64, 256, 0, stream>>>(o_A, output, eye, stomach, o_state);

  k_grad<<<BN_, 128, 0, stream>>>(state, o_out, o_state, ws_q, ws_k,
                                  eb1, eb2, eb3, o_eb1, o_eb2, o_eb3,
                                  ws_p, ws_ns);

  k_piece<<<BN_, 256, 0, stream>>>(w1, g1, ws_p + 0 * 131072, ws_ns + 0 * 131072,
                                   stepc, o_w1, o_g1);
  k_piece<<<BN_, 256, 0, stream>>>(w2, g2, ws_p + 1 * 131072, ws_ns + 1 * 131072,
                                   stepc, o_w2, o_g2);
  k_piece<<<BN_, 256, 0, stream>>>(w3, g3, ws_p + 2 * 131072, ws_ns + 2 * 131072,
                                   stepc, o_w3, o_g3);
}